// GNNModel_72627896975415
// MI455X (gfx1250) — compile-verified
//
#include <hip/hip_runtime.h>

typedef __attribute__((ext_vector_type(2))) float v2f;
typedef __attribute__((ext_vector_type(8))) float v8f;

#define GCN_N 50000
#define GCN_E 800000

__device__ __forceinline__ void atomic_add_f32(float* p, float v) {
  __hip_atomic_fetch_add(p, v, __ATOMIC_RELAXED, __HIP_MEMORY_SCOPE_AGENT);
}

// ---------------- degree + norm ----------------
__global__ void gcn_degree_kernel(const int* __restrict__ src,
                                  const int* __restrict__ dst,
                                  unsigned* __restrict__ deg_out,
                                  unsigned* __restrict__ deg_in,
                                  int n_edges) {
  int e = blockIdx.x * blockDim.x + threadIdx.x;
  if (e < n_edges) {
    atomicAdd(&deg_out[src[e]], 1u);
    atomicAdd(&deg_in[dst[e]], 1u);
  }
}

__global__ void gcn_norm_kernel(const unsigned* __restrict__ deg_out,
                                const unsigned* __restrict__ deg_in,
                                float* __restrict__ norm_out,
                                float* __restrict__ norm_in,
                                int n_nodes) {
  int n = blockIdx.x * blockDim.x + threadIdx.x;
  if (n >= n_nodes) return;
  unsigned dout = deg_out[n]; if (dout < 1u) dout = 1u;
  unsigned din  = deg_in[n];  if (din  < 1u) din  = 1u;
  norm_out[n] = rsqrtf((float)dout);
  norm_in[n]  = rsqrtf((float)din);
}

// ---------------- y[n,0:64] = x[n,0:64] * rownorm[n] ----------------
__global__ void gcn_scale_rows64_kernel(const float* __restrict__ x,
                                        const float* __restrict__ rownorm,
                                        float* __restrict__ y,
                                        int n_nodes) {
  int i = blockIdx.x * blockDim.x + threadIdx.x;  // over n_nodes*16 float4s
  if (i >= n_nodes * 16) return;
  int n = i >> 4;
  float s = rownorm[n];
  float4 v = ((const float4*)x)[i];
  v.x *= s; v.y *= s; v.z *= s; v.w *= s;
  ((float4*)y)[i] = v;
}

// ---------------- scatter-add of 64-wide rows: agg[dst] += x[src] ----------------
__global__ void gcn_scatter64_kernel(const int* __restrict__ src,
                                     const int* __restrict__ dst,
                                     const float* __restrict__ x,
                                     float* __restrict__ agg,
                                     int n_edges) {
  long long gid = (long long)blockIdx.x * blockDim.x + threadIdx.x;
  if (gid >= (long long)n_edges * 16) return;
  int e = (int)(gid >> 4);
  int q = (int)(gid & 15);
  int s = src[e];
  int d = dst[e];
  float4 v = ((const float4*)(x + (size_t)s * 64))[q];
  float* p = agg + (size_t)d * 64 + (size_t)q * 4;
  atomic_add_f32(p + 0, v.x);
  atomic_add_f32(p + 1, v.y);
  atomic_add_f32(p + 2, v.z);
  atomic_add_f32(p + 3, v.w);
}

// ---------------- layer-1 GEMM (WMMA f32 16x16x4) ----------------
// Y[row, col] = relu( ((norm_in[row]*A[row,:]) @ W)[col] + bias[col] ) * norm_out[row]
// A: [N,64], W: [64,128], Y: [N,128]. One wave32 per 16x16 output tile.
__global__ void __launch_bounds__(256)
gcn_gemm1_kernel(const float* __restrict__ A,
                 const float* __restrict__ W,
                 const float* __restrict__ bias,
                 const float* __restrict__ norm_in,
                 const float* __restrict__ norm_out,
                 float* __restrict__ Y,
                 int n_tiles_m) {
  int wave = threadIdx.x >> 5;
  int tile = blockIdx.x * 8 + wave;   // mt*8 + nt, 8 N-tiles (128/16)
  int mt = tile >> 3;
  int nt = tile & 7;
  if (mt >= n_tiles_m) return;
  int lane = threadIdx.x & 31;
  int half = lane >> 4;   // 0: K pair {0,1}, 1: K pair {2,3}
  int l16  = lane & 15;
  int arow = mt * 16 + l16;
  int bcol = nt * 16 + l16;
  float rs = norm_in[arow];
  const float* ap = A + (size_t)arow * 64;
  v8f acc = {};
#pragma unroll
  for (int k0 = 0; k0 < 64; k0 += 4) {
    int ka = k0 + half * 2;
    v2f a;
    a.x = ap[ka] * rs;
    a.y = ap[ka + 1] * rs;
    v2f b;
    b.x = W[(size_t)ka * 128 + bcol];
    b.y = W[(size_t)(ka + 1) * 128 + bcol];
    acc = __builtin_amdgcn_wmma_f32_16x16x4_f32(
        false, a, false, b, (short)0, acc, false, false);
  }
  float bb = bias[bcol];
#pragma unroll
  for (int r = 0; r < 8; ++r) {
    int row = mt * 16 + half * 8 + r;
    float v = fmaxf(acc[r] + bb, 0.0f) * norm_out[row];
    Y[(size_t)row * 128 + bcol] = v;
  }
}

// ---------------- layer-2 pre-aggregation GEMM (WMMA f32 16x16x4) ----------------
// Y = A @ W.  A: [N,128] (already row-scaled), W: [128,64], Y: [N,64].
__global__ void __launch_bounds__(256)
gcn_gemm2_kernel(const float* __restrict__ A,
                 const float* __restrict__ W,
                 float* __restrict__ Y,
                 int n_tiles) {
  int wave = threadIdx.x >> 5;
  int tile = blockIdx.x * 8 + wave;   // mt*4 + nt, 4 N-tiles (64/16)
  if (tile >= n_tiles) return;
  int mt = tile >> 2;
  int nt = tile & 3;
  int lane = threadIdx.x & 31;
  int half = lane >> 4;
  int l16  = lane & 15;
  int arow = mt * 16 + l16;
  int bcol = nt * 16 + l16;
  const float* ap = A + (size_t)arow * 128;
  v8f acc = {};
#pragma unroll
  for (int k0 = 0; k0 < 128; k0 += 4) {
    int ka = k0 + half * 2;
    v2f a;
    a.x = ap[ka];
    a.y = ap[ka + 1];
    v2f b;
    b.x = W[(size_t)ka * 64 + bcol];
    b.y = W[(size_t)(ka + 1) * 64 + bcol];
    acc = __builtin_amdgcn_wmma_f32_16x16x4_f32(
        false, a, false, b, (short)0, acc, false, false);
  }
#pragma unroll
  for (int r = 0; r < 8; ++r) {
    int row = mt * 16 + half * 8 + r;
    Y[(size_t)row * 64 + bcol] = acc[r];
  }
}

// ---------------- final: out[n,f] = agg2[n,f]*norm_in[n] + b2[f] ----------------
__global__ void gcn_final_kernel(const float* __restrict__ agg2,
                                 const float* __restrict__ norm_in,
                                 const float* __restrict__ bias,
                                 float* __restrict__ out,
                                 int n_nodes) {
  int i = blockIdx.x * blockDim.x + threadIdx.x;  // n_nodes*16 float4s
  if (i >= n_nodes * 16) return;
  int n = i >> 4;
  int q = i & 15;
  float s = norm_in[n];
  float4 v = ((const float4*)agg2)[i];
  float4 bb = ((const float4*)bias)[q];
  v.x = v.x * s + bb.x;
  v.y = v.y * s + bb.y;
  v.z = v.z * s + bb.z;
  v.w = v.w * s + bb.w;
  ((float4*)out)[i] = v;
}

extern "C" void kernel_launch(void* const* d_in, const int* in_sizes, int n_in,
                              void* d_out, int out_size, void* d_ws, size_t ws_size,
                              hipStream_t stream) {
  (void)in_sizes; (void)n_in; (void)out_size; (void)ws_size;
  const float* h   = (const float*)d_in[0];
  const int*   src = (const int*)d_in[1];
  const int*   dst = (const int*)d_in[2];
  const float* W1  = (const float*)d_in[3];
  const float* b1  = (const float*)d_in[4];
  const float* W2  = (const float*)d_in[5];
  const float* b2  = (const float*)d_in[6];
  float* out = (float*)d_out;

  const int N = GCN_N;
  const int E = GCN_E;

  // workspace carve-out (256B aligned slabs)
  char* ws = (char*)d_ws;
  size_t off = 0;
  auto carve = [&](size_t bytes) -> void* {
    void* p = ws + off;
    off += (bytes + 255) & ~(size_t)255;
    return p;
  };
  unsigned* deg_out = (unsigned*)carve((size_t)N * 4);
  unsigned* deg_in  = (unsigned*)carve((size_t)N * 4);
  float* norm_out   = (float*)carve((size_t)N * 4);
  float* norm_in    = (float*)carve((size_t)N * 4);
  float* hs         = (float*)carve((size_t)N * 64 * 4);   // h * norm_out
  float* agg1       = (float*)carve((size_t)N * 64 * 4);   // layer-1 aggregate
  float* h1s        = (float*)carve((size_t)N * 128 * 4);  // relu(...)*norm_out
  float* t2         = (float*)carve((size_t)N * 64 * 4);   // h1s @ W2
  float* agg2       = (float*)carve((size_t)N * 64 * 4);   // layer-2 aggregate

  // zero accumulators each call (graph-capture safe memset nodes)
  hipMemsetAsync(deg_out, 0, (size_t)N * 4, stream);
  hipMemsetAsync(deg_in,  0, (size_t)N * 4, stream);
  hipMemsetAsync(agg1,    0, (size_t)N * 64 * 4, stream);
  hipMemsetAsync(agg2,    0, (size_t)N * 64 * 4, stream);

  gcn_degree_kernel<<<(E + 255) / 256, 256, 0, stream>>>(src, dst, deg_out, deg_in, E);
  gcn_norm_kernel<<<(N + 255) / 256, 256, 0, stream>>>(deg_out, deg_in, norm_out, norm_in, N);

  // layer 1: aggregate 64-dim messages, then WMMA GEMM with fused bias/relu/norms
  gcn_scale_rows64_kernel<<<(N * 16 + 255) / 256, 256, 0, stream>>>(h, norm_out, hs, N);
  {
    long long thr = (long long)E * 16;
    gcn_scatter64_kernel<<<(unsigned)((thr + 255) / 256), 256, 0, stream>>>(src, dst, hs, agg1, E);
  }
  gcn_gemm1_kernel<<<N / 16, 256, 0, stream>>>(agg1, W1, b1, norm_in, norm_out, h1s, N / 16);

  // layer 2: apply W2 BEFORE aggregation (64-dim edge traffic instead of 128)
  {
    int tiles = (N / 16) * 4;
    gcn_gemm2_kernel<<<(tiles + 7) / 8, 256, 0, stream>>>(h1s, W2, t2, tiles);
  }
  {
    long long thr = (long long)E * 16;
    gcn_scatter64_kernel<<<(unsigned)((thr + 255) / 256), 256, 0, stream>>>(src, dst, t2, agg2, E);
  }
  gcn_final_kernel<<<(N * 16 + 255) / 256, 256, 0, stream>>>(agg2, norm_in, b2, out, N);
}